// LearnedBackprop_80058190397441
// MI455X (gfx1250) — compile-verified
//
#include <hip/hip_runtime.h>
#include <math.h>

typedef __attribute__((ext_vector_type(2))) float v2f;
typedef __attribute__((ext_vector_type(4))) float v4f;
typedef __attribute__((ext_vector_type(8))) float v8f;

#define TPB 256
#define NBLK 4096
#define HIDDEN 32
#define KPAD 44   // 10 features + 32 hidden, padded to multiple of 4
#define EPS 1e-8f

__device__ __forceinline__ float sigmoidf(float x) {
    return 1.0f / (1.0f + expf(-x));
}

// ---------------------------------------------------------------------------
// Kernel 1: streaming reduction over grad/param (memory-bound, ~268 MB once).
// Non-temporal b128 loads; deterministic fixed-grid partial sums (no atomics).
// ---------------------------------------------------------------------------
__global__ __launch_bounds__(TPB) void reduce_kernel(
    const float* __restrict__ grad, const float* __restrict__ param,
    float* __restrict__ partials, int n4) {
    const v4f* g4 = (const v4f*)grad;
    const v4f* p4 = (const v4f*)param;
    int tid = blockIdx.x * TPB + threadIdx.x;
    int stride = gridDim.x * TPB;

    float sgg = 0.0f, spp = 0.0f, sgp = 0.0f, slg = 0.0f;
    for (int i = tid; i < n4; i += stride) {
        v4f gv = __builtin_nontemporal_load(g4 + i);
        v4f pv = __builtin_nontemporal_load(p4 + i);
#pragma unroll
        for (int k = 0; k < 4; ++k) {
            float g = gv[k];
            float p = pv[k];
            sgg = fmaf(g, g, sgg);
            spp = fmaf(p, p, spp);
            sgp = fmaf(g, p, sgp);
            slg += copysignf(log1pf(fabsf(g)), g);
        }
    }

    __shared__ float red[4 * TPB];
    red[0 * TPB + threadIdx.x] = sgg;
    red[1 * TPB + threadIdx.x] = spp;
    red[2 * TPB + threadIdx.x] = sgp;
    red[3 * TPB + threadIdx.x] = slg;
    __syncthreads();
    for (int s = TPB / 2; s > 0; s >>= 1) {
        if (threadIdx.x < s) {
            red[0 * TPB + threadIdx.x] += red[0 * TPB + threadIdx.x + s];
            red[1 * TPB + threadIdx.x] += red[1 * TPB + threadIdx.x + s];
            red[2 * TPB + threadIdx.x] += red[2 * TPB + threadIdx.x + s];
            red[3 * TPB + threadIdx.x] += red[3 * TPB + threadIdx.x + s];
        }
        __syncthreads();
    }
    if (threadIdx.x == 0) {
        v4f o;
        o.x = red[0 * TPB];
        o.y = red[1 * TPB];
        o.z = red[2 * TPB];
        o.w = red[3 * TPB];
        ((v4f*)partials)[blockIdx.x] = o;
    }
}

// ---------------------------------------------------------------------------
// Kernel 2: finish reductions, build features, run LSTM cell + heads.
// Gates GEMV (128 x 44) done with V_WMMA_F32_16X16X4_F32 on wave 0.
// ---------------------------------------------------------------------------
__global__ __launch_bounds__(TPB) void final_kernel(
    const float* __restrict__ partials, int nblk, float inv_n,
    const float* __restrict__ neuromod, const float* __restrict__ comp_delta,
    const float* __restrict__ m_prev, const float* __restrict__ h_prev,
    const float* __restrict__ c_prev,
    const float* __restrict__ W_ih, const float* __restrict__ W_hh,
    const float* __restrict__ b_ih, const float* __restrict__ b_hh,
    const float* __restrict__ Ws, const float* __restrict__ bs,
    const float* __restrict__ Wg1, const float* __restrict__ bg1,
    const float* __restrict__ Wg2, const float* __restrict__ bg2,
    float* __restrict__ out) {
    __shared__ float red[4 * TPB];
    __shared__ float Wc[128 * KPAD];   // [W_ih | W_hh | 0 0], row-major, K padded
    __shared__ float xs[KPAD];         // [features(10) ; h_prev(32) ; 0 0]
    __shared__ float gates[128];
    __shared__ float hnew[HIDDEN];
    __shared__ float hid[HIDDEN / 2];

    // --- stage combined weight matrix into LDS ---
    for (int idx = threadIdx.x; idx < 128 * KPAD; idx += TPB) {
        int row = idx / KPAD;
        int k = idx - row * KPAD;
        float v;
        if (k < 10)       v = W_ih[row * 10 + k];
        else if (k < 42)  v = W_hh[row * 32 + (k - 10)];
        else              v = 0.0f;
        Wc[idx] = v;
    }

    // --- reduce per-block partials (deterministic order) ---
    float a0 = 0.0f, a1 = 0.0f, a2 = 0.0f, a3 = 0.0f;
    const v4f* pp = (const v4f*)partials;
    for (int i = threadIdx.x; i < nblk; i += TPB) {
        v4f v = pp[i];
        a0 += v.x; a1 += v.y; a2 += v.z; a3 += v.w;
    }
    red[0 * TPB + threadIdx.x] = a0;
    red[1 * TPB + threadIdx.x] = a1;
    red[2 * TPB + threadIdx.x] = a2;
    red[3 * TPB + threadIdx.x] = a3;
    __syncthreads();
    for (int s = TPB / 2; s > 0; s >>= 1) {
        if (threadIdx.x < s) {
            red[0 * TPB + threadIdx.x] += red[0 * TPB + threadIdx.x + s];
            red[1 * TPB + threadIdx.x] += red[1 * TPB + threadIdx.x + s];
            red[2 * TPB + threadIdx.x] += red[2 * TPB + threadIdx.x + s];
            red[3 * TPB + threadIdx.x] += red[3 * TPB + threadIdx.x + s];
        }
        __syncthreads();
    }

    // --- build feature vector xs[0..43] ---
    if (threadIdx.x < HIDDEN) xs[10 + threadIdx.x] = h_prev[threadIdx.x];
    if (threadIdx.x < 4)      xs[6 + threadIdx.x]  = neuromod[threadIdx.x];
    if (threadIdx.x == 0) {
        float sgg = red[0 * TPB], spp = red[1 * TPB];
        float sgp = red[2 * TPB], slg = red[3 * TPB];
        float g_norm = fmaxf(sqrtf(sgg), EPS);
        float p_norm = fmaxf(sqrtf(spp), EPS);
        xs[0] = logf(g_norm + EPS);
        xs[1] = logf(p_norm + EPS);
        xs[2] = sgp / (g_norm * p_norm + EPS);
        xs[3] = slg * inv_n;
        xs[4] = 0.9f * m_prev[0] + 0.1f * g_norm;
        xs[5] = comp_delta[0];
        xs[42] = 0.0f;
        xs[43] = 0.0f;
    }
    __syncthreads();

    // --- gates = Wc(128x44) @ xs(44) via WMMA f32 16x16x4 on wave 0 ---
    // A fragment (16x4 f32): lanes 0-15 hold K={k0,k0+1}, lanes 16-31 K={k0+2,k0+3},
    // M = lane & 15.  B broadcast xs[k] across all 16 columns, so every column of
    // D equals the GEMV result; lane 0 holds rows M=0..7, lane 16 rows M=8..15.
    if (threadIdx.x < 32) {
        int lane = threadIdx.x;
        int mrow = lane & 15;
        int khalf = (lane < 16) ? 0 : 2;
#pragma unroll 1
        for (int chunk = 0; chunk < 8; ++chunk) {
            int row = chunk * 16 + mrow;
            v8f acc = {0.f, 0.f, 0.f, 0.f, 0.f, 0.f, 0.f, 0.f};
            for (int k0 = 0; k0 < KPAD; k0 += 4) {
                int ka = k0 + khalf;
                v2f a, b;
                a.x = Wc[row * KPAD + ka];
                a.y = Wc[row * KPAD + ka + 1];
                b.x = xs[ka];
                b.y = xs[ka + 1];
                acc = __builtin_amdgcn_wmma_f32_16x16x4_f32(
                    /*neg_a=*/false, a, /*neg_b=*/false, b,
                    /*c_mod=*/(short)0, acc, /*reuse_a=*/false, /*reuse_b=*/false);
            }
            if (mrow == 0) {  // lanes 0 and 16 hold column N=0
                int base = chunk * 16 + ((lane < 16) ? 0 : 8);
#pragma unroll
                for (int r = 0; r < 8; ++r) gates[base + r] = acc[r];
            }
        }
    }
    __syncthreads();

    // --- LSTM elementwise (torch gate order i,f,g,o) ---
    if (threadIdx.x < HIDDEN) {
        int j = threadIdx.x;
        float iv = gates[j]      + b_ih[j]      + b_hh[j];
        float fv = gates[32 + j] + b_ih[32 + j] + b_hh[32 + j];
        float gv = gates[64 + j] + b_ih[64 + j] + b_hh[64 + j];
        float ov = gates[96 + j] + b_ih[96 + j] + b_hh[96 + j];
        float c = sigmoidf(fv) * c_prev[j] + sigmoidf(iv) * tanhf(gv);
        hnew[j] = sigmoidf(ov) * tanhf(c);
    }
    __syncthreads();

    // --- gate head: hid = tanh(Wg1 @ h + bg1) ---
    if (threadIdx.x < HIDDEN / 2) {
        float acc = bg1[threadIdx.x];
        for (int j = 0; j < HIDDEN; ++j)
            acc = fmaf(Wg1[threadIdx.x * HIDDEN + j], hnew[j], acc);
        hid[threadIdx.x] = tanhf(acc);
    }
    __syncthreads();

    // --- scale head + combine + clip ---
    if (threadIdx.x == 0) {
        float rs = bs[0];
        for (int j = 0; j < HIDDEN; ++j) rs = fmaf(Ws[j], hnew[j], rs);
        float scale = expf(tanhf(rs));
        float cg = bg2[0];
        for (int k = 0; k < HIDDEN / 2; ++k) cg = fmaf(Wg2[k], hid[k], cg);
        cg = sigmoidf(cg);
        float mult = scale * (0.5f + cg);
        out[0] = fminf(fmaxf(mult, 0.01f), 10.0f);
    }
}

extern "C" void kernel_launch(void* const* d_in, const int* in_sizes, int n_in,
                              void* d_out, int out_size, void* d_ws, size_t ws_size,
                              hipStream_t stream) {
    const float* grad       = (const float*)d_in[0];
    const float* param      = (const float*)d_in[1];
    const float* neuromod   = (const float*)d_in[2];
    const float* comp_delta = (const float*)d_in[3];
    const float* m_prev     = (const float*)d_in[4];
    const float* h_prev     = (const float*)d_in[5];
    const float* c_prev     = (const float*)d_in[6];
    const float* W_ih       = (const float*)d_in[7];
    const float* W_hh       = (const float*)d_in[8];
    const float* b_ih       = (const float*)d_in[9];
    const float* b_hh       = (const float*)d_in[10];
    const float* Ws         = (const float*)d_in[11];
    const float* bs         = (const float*)d_in[12];
    const float* Wg1        = (const float*)d_in[13];
    const float* bg1        = (const float*)d_in[14];
    const float* Wg2        = (const float*)d_in[15];
    const float* bg2        = (const float*)d_in[16];

    int n = in_sizes[0];     // 4096 * 8192
    int n4 = n >> 2;
    float* partials = (float*)d_ws;  // NBLK * 4 floats = 64 KB

    reduce_kernel<<<NBLK, TPB, 0, stream>>>(grad, param, partials, n4);
    final_kernel<<<1, TPB, 0, stream>>>(
        partials, NBLK, 1.0f / (float)n, neuromod, comp_delta,
        m_prev, h_prev, c_prev, W_ih, W_hh, b_ih, b_hh,
        Ws, bs, Wg1, bg1, Wg2, bg2, (float*)d_out);
}